// WRF_46196668235945
// MI455X (gfx1250) — compile-verified
//
#include <hip/hip_runtime.h>
#include <math.h>

// ---------------- problem geometry / physics constants ----------------
constexpr int GNX = 384, GNY = 384, GNZ = 48;
constexpr float DXf = 1000.0f, DYf = 1000.0f;
constexpr float DZf = 1.0f / 49.0f;          // 1/(NZ+1)
constexpr float PREFf = 100000.0f;
constexpr float RDf = 287.0f;
constexpr float GGf = 9.81f;

constexpr int SZ_C = GNZ * GNY * GNX;           // 7,077,888  (Theta, Mu, alpha, p)
constexpr int SZ_U = GNZ * GNY * (GNX + 1);     // 7,096,320
constexpr int SZ_V = GNZ * (GNY + 1) * GNX;     // 7,096,320
constexpr int SZ_W = (GNZ - 1) * GNY * GNX;     // 6,930,432  (W, Phi)
constexpr int BLK  = 256;

// ---------------- gfx1250 async global->LDS path (probe-confirmed to lower
// to global_load_async_to_lds_b32 + s_wait_asynccnt on this toolchain) ------
#if defined(__HIP_DEVICE_COMPILE__) && defined(__gfx1250__) && \
    __has_builtin(__builtin_amdgcn_global_load_async_to_lds_b32) && \
    __has_builtin(__builtin_amdgcn_s_wait_asynccnt)
#define USE_ASYNC_LDS 1
#else
#define USE_ASYNC_LDS 0
#endif

#if USE_ASYNC_LDS
typedef __attribute__((address_space(1))) int* gint_ptr;
typedef __attribute__((address_space(3))) int* lint_ptr;
// issue one per-lane 4B async DMA: global[src] -> LDS slot
#define ASYNC_STAGE(srcp, slotp) \
  __builtin_amdgcn_global_load_async_to_lds_b32((gint_ptr)(srcp), (lint_ptr)(slotp), 0, 0)
#define ASYNC_WAIT() \
  do { __builtin_amdgcn_s_wait_asynccnt(0); asm volatile("" ::: "memory"); } while (0)
#endif

// ---------------- indexing helpers ----------------
__device__ __forceinline__ int idxC(int k, int j, int i) { return (k * GNY + j) * GNX + i; }
__device__ __forceinline__ int idxU(int k, int j, int i) { return (k * GNY + j) * (GNX + 1) + i; }
__device__ __forceinline__ int idxV(int k, int j, int i) { return (k * (GNY + 1) + j) * GNX + i; }
__device__ __forceinline__ int idxS(int j, int i)        { return j * GNX + i; }

// u at x-face f in [0,385):  U[f] / (0.5*(Mu[(f-1)%384] + Mu[f%384]))
__device__ __forceinline__ float uFace(const float* U, const float* Mu, int k, int j, int f) {
  int xc = (f == GNX) ? 0 : f;
  int xm = (f == 0 || f == GNX) ? GNX - 1 : f - 1;
  float mux = 0.5f * (Mu[idxC(k, j, xm)] + Mu[idxC(k, j, xc)]);
  return U[idxU(k, j, f)] / mux;
}

// v at y-face r in [0,385)
__device__ __forceinline__ float vFace(const float* V, const float* Mu, int k, int r, int i) {
  int yc = (r == GNY) ? 0 : r;
  int ym = (r == 0 || r == GNY) ? GNY - 1 : r - 1;
  float muy = 0.5f * (Mu[idxC(k, ym, i)] + Mu[idxC(k, yc, i)]);
  return V[idxV(k, r, i)] / muy;
}

// w at z-face k in [0,47)
__device__ __forceinline__ float wVal(const float* W, const float* Mu, int k, int j, int i) {
  float muz = 0.5f * (Mu[idxC(k, j, i)] + Mu[idxC(k + 1, j, i)]);
  return W[idxC(k, j, i)] / muz;
}

// Omega at z-face k in [0,47): mirrors reference op order exactly
// omega = -w*G/bar_z(alpha)/bar_z(Mu); Omega = omega*bar_z(Mu)
__device__ __forceinline__ float OmegaVal(const float* W, const float* Mu, const float* alpha,
                                          int k, int j, int i) {
  int a = idxC(k, j, i), b = idxC(k + 1, j, i);
  float muz = 0.5f * (Mu[a] + Mu[b]);
  float az  = 0.5f * (alpha[a] + alpha[b]);
  float w   = W[a] / muz;
  float om  = -w * GGf / az / muz;
  return om * muz;
}

// zero-padded Omega: index m in [0,48]; 0 at m==0 and m==48
__device__ __forceinline__ float OmegaPad(const float* W, const float* Mu, const float* alpha,
                                          int m, int j, int i) {
  if (m == 0 || m == GNZ) return 0.0f;
  return OmegaVal(W, Mu, alpha, m - 1, j, i);
}

// pad_z(p, P_t, P_s): m in [0,49]
__device__ __forceinline__ float pPad(const float* p, const float* P_t, const float* P_s,
                                      int m, int j, int i) {
  if (m == 0) return P_t[idxS(j, i)];
  if (m == GNZ + 1) return P_s[idxS(j, i)];
  return p[idxC(m - 1, j, i)];
}

// pad_z(Phi, Phi_t, Phi_s): m in [0,48]  (Phi has 47 levels)
__device__ __forceinline__ float phiPad(const float* Phi, const float* Phi_t, const float* Phi_s,
                                        int m, int j, int i) {
  if (m == 0) return Phi_t[idxS(j, i)];
  if (m == GNZ) return Phi_s[idxS(j, i)];
  return Phi[idxC(m - 1, j, i)];
}

// ---------------- kernel: derived fields alpha, p ----------------
__global__ void k_derive(const float* __restrict__ Theta, const float* __restrict__ Mu,
                         const float* __restrict__ Phi, const float* __restrict__ Phi_t,
                         const float* __restrict__ Phi_s,
                         float* __restrict__ alpha, float* __restrict__ p) {
  int tid = threadIdx.x;
  int idx = blockIdx.x * BLK + tid;
  if (idx >= SZ_C) return;
#if USE_ASYNC_LDS
  __shared__ float sTh[BLK];
  __shared__ float sMu[BLK];
  // async-DMA the coalesced center loads into LDS; overlaps with Phi gathers
  ASYNC_STAGE(Theta + idx, &sTh[tid]);
  ASYNC_STAGE(Mu + idx, &sMu[tid]);
#endif
  int i = idx % GNX;
  int t = idx / GNX;
  int j = t % GNY;
  int k = t / GNY;
  int s = idxS(j, i);
  float phiLo = (k == 0) ? Phi_t[s] : Phi[idxC(k - 1, j, i)];
  float phiHi = (k == GNZ - 1) ? Phi_s[s] : Phi[idxC(k, j, i)];
#if USE_ASYNC_LDS
  ASYNC_WAIT();
  float mu = sMu[tid];
  float th = sTh[tid] / mu;
#else
  float mu = Mu[idx];
  float th = Theta[idx] / mu;
#endif
  float a = -((phiHi - phiLo) / DZf) / mu;
  alpha[idx] = a;
  p[idx] = PREFf * powf(RDf * th / PREFf / a, 1.4f);
}

// ---------------- kernel: R_U and U update ----------------
__global__ void k_ru(const float* __restrict__ Ub,
                     const float* __restrict__ U, const float* __restrict__ V,
                     const float* __restrict__ W, const float* __restrict__ Mu,
                     const float* __restrict__ Phi, const float* __restrict__ Phi_t,
                     const float* __restrict__ Phi_s, const float* __restrict__ P_t,
                     const float* __restrict__ P_s,
                     const float* __restrict__ alpha, const float* __restrict__ p,
                     const int* __restrict__ dtp, float cfac,
                     float* __restrict__ Uout) {
  int idx = blockIdx.x * blockDim.x + threadIdx.x;
  if (idx >= SZ_U) return;
#if USE_ASYNC_LDS
  __shared__ float sBase[BLK];
  ASYNC_STAGE(Ub + idx, &sBase[threadIdx.x]);   // hide base load behind gathers
#endif
  int c = idx % (GNX + 1);
  int t = idx / (GNX + 1);
  int j = t % GNY;
  int k = t / GNY;

  int xc = (c == GNX) ? 0 : c;
  int xm = (c == 0 || c == GNX) ? GNX - 1 : c - 1;
  int jm = (j == 0) ? GNY - 1 : j - 1;
  int jp = (j == GNY - 1) ? 0 : j + 1;

  // faces for padU/padu at positions c, c+1, c+2 of the 387-wide pad
  int fA = (c == 0) ? GNX : c - 1;
  int fB = c;
  int fC = (c == GNX) ? 0 : c + 1;

  float UA = U[idxU(k, j, fA)], UBv = U[idxU(k, j, fB)], UCv = U[idxU(k, j, fC)];
  float uA = uFace(U, Mu, k, j, fA);
  float uB = uFace(U, Mu, k, j, fB);
  float uC = uFace(U, Mu, k, j, fC);
  float term1 = -((0.5f * (UBv + UCv)) * (0.5f * (uB + uC)) -
                  (0.5f * (UA + UBv)) * (0.5f * (uA + uB))) / DXf;

  float u_jm = uFace(U, Mu, k, jm, fB);
  float u_jp = uFace(U, Mu, k, jp, fB);
  float BxV_j  = 0.5f * (V[idxV(k, j,     xm)] + V[idxV(k, j,     xc)]);
  float BxV_j1 = 0.5f * (V[idxV(k, j + 1, xm)] + V[idxV(k, j + 1, xc)]);
  float term2 = -(BxV_j1 * (0.5f * (uB + u_jp)) - BxV_j * (0.5f * (u_jm + uB))) / DYf;

  float u_km = (k > 0) ? uFace(U, Mu, k - 1, j, fB) : 0.0f;
  float u_kp = (k < GNZ - 1) ? uFace(U, Mu, k + 1, j, fB) : 0.0f;
  float uz_k  = 0.5f * (u_km + uB);
  float uz_k1 = 0.5f * (uB + u_kp);
  float Ox_k  = 0.5f * (OmegaPad(W, Mu, alpha, k,     j, xm) + OmegaPad(W, Mu, alpha, k,     j, xc));
  float Ox_k1 = 0.5f * (OmegaPad(W, Mu, alpha, k + 1, j, xm) + OmegaPad(W, Mu, alpha, k + 1, j, xc));
  float term3 = -(Ox_k1 * uz_k1 - Ox_k * uz_k) / DZf;

  float muX = 0.5f * (Mu[idxC(k, j, xm)] + Mu[idxC(k, j, xc)]);
  float aX  = 0.5f * (alpha[idxC(k, j, xm)] + alpha[idxC(k, j, xc)]);
  float dpx = (p[idxC(k, j, xc)] - p[idxC(k, j, xm)]) / DXf;

  float pz_k_m  = 0.5f * (pPad(p, P_t, P_s, k,     j, xm) + pPad(p, P_t, P_s, k + 1, j, xm));
  float pz_k_c  = 0.5f * (pPad(p, P_t, P_s, k,     j, xc) + pPad(p, P_t, P_s, k + 1, j, xc));
  float pz_k1_m = 0.5f * (pPad(p, P_t, P_s, k + 1, j, xm) + pPad(p, P_t, P_s, k + 2, j, xm));
  float pz_k1_c = 0.5f * (pPad(p, P_t, P_s, k + 1, j, xc) + pPad(p, P_t, P_s, k + 2, j, xc));
  float dpzx = (0.5f * (pz_k1_m + pz_k1_c) - 0.5f * (pz_k_m + pz_k_c)) / DZf;

  float phz_m = 0.5f * (phiPad(Phi, Phi_t, Phi_s, k, j, xm) + phiPad(Phi, Phi_t, Phi_s, k + 1, j, xm));
  float phz_c = 0.5f * (phiPad(Phi, Phi_t, Phi_s, k, j, xc) + phiPad(Phi, Phi_t, Phi_s, k + 1, j, xc));
  float dphx = (phz_c - phz_m) / DXf;

  float R = term1 + term2 + term3 - muX * aX * dpx - dpzx * dphx;
  float coef = cfac * (float)dtp[0];
#if USE_ASYNC_LDS
  ASYNC_WAIT();
  float base = sBase[threadIdx.x];
#else
  float base = Ub[idx];
#endif
  Uout[idx] = base + coef * R;
}

// ---------------- kernel: R_V and V update ----------------
__global__ void k_rv(const float* __restrict__ Vb,
                     const float* __restrict__ U, const float* __restrict__ V,
                     const float* __restrict__ W, const float* __restrict__ Mu,
                     const float* __restrict__ Phi, const float* __restrict__ Phi_t,
                     const float* __restrict__ Phi_s, const float* __restrict__ P_t,
                     const float* __restrict__ P_s,
                     const float* __restrict__ alpha, const float* __restrict__ p,
                     const int* __restrict__ dtp, float cfac,
                     float* __restrict__ Vout) {
  int idx = blockIdx.x * blockDim.x + threadIdx.x;
  if (idx >= SZ_V) return;
#if USE_ASYNC_LDS
  __shared__ float sBase[BLK];
  ASYNC_STAGE(Vb + idx, &sBase[threadIdx.x]);
#endif
  int i = idx % GNX;
  int t = idx / GNX;
  int r = t % (GNY + 1);
  int k = t / (GNY + 1);

  int rc = (r == GNY) ? 0 : r;
  int rm = (r == 0 || r == GNY) ? GNY - 1 : r - 1;
  int xm = (i == 0) ? GNX - 1 : i - 1;
  int xp = (i == GNX - 1) ? 0 : i + 1;

  float ByU_i  = 0.5f * (U[idxU(k, rm, i)]     + U[idxU(k, rc, i)]);
  float ByU_i1 = 0.5f * (U[idxU(k, rm, i + 1)] + U[idxU(k, rc, i + 1)]);
  float v_xm = vFace(V, Mu, k, r, xm);
  float vB   = vFace(V, Mu, k, r, i);
  float v_xp = vFace(V, Mu, k, r, xp);
  float term1 = -(ByU_i1 * (0.5f * (vB + v_xp)) - ByU_i * (0.5f * (v_xm + vB))) / DXf;

  int rA  = (r == 0) ? GNY : r - 1;
  int rC2 = (r == GNY) ? 0 : r + 1;
  float VA = V[idxV(k, rA, i)], VB = V[idxV(k, r, i)], VC = V[idxV(k, rC2, i)];
  float vA = vFace(V, Mu, k, rA, i);
  float vC = vFace(V, Mu, k, rC2, i);
  float term2 = -((0.5f * (VB + VC)) * (0.5f * (vB + vC)) -
                  (0.5f * (VA + VB)) * (0.5f * (vA + vB))) / DYf;

  float v_km = (k > 0) ? vFace(V, Mu, k - 1, r, i) : 0.0f;
  float v_kp = (k < GNZ - 1) ? vFace(V, Mu, k + 1, r, i) : 0.0f;
  float vz_k  = 0.5f * (v_km + vB);
  float vz_k1 = 0.5f * (vB + v_kp);
  float Oy_k  = 0.5f * (OmegaPad(W, Mu, alpha, k,     rm, i) + OmegaPad(W, Mu, alpha, k,     rc, i));
  float Oy_k1 = 0.5f * (OmegaPad(W, Mu, alpha, k + 1, rm, i) + OmegaPad(W, Mu, alpha, k + 1, rc, i));
  float term3 = -(Oy_k1 * vz_k1 - Oy_k * vz_k) / DZf;

  float muY = 0.5f * (Mu[idxC(k, rm, i)] + Mu[idxC(k, rc, i)]);
  float aY  = 0.5f * (alpha[idxC(k, rm, i)] + alpha[idxC(k, rc, i)]);
  float dpy = (p[idxC(k, rc, i)] - p[idxC(k, rm, i)]) / DYf;

  float pz_k_m  = 0.5f * (pPad(p, P_t, P_s, k,     rm, i) + pPad(p, P_t, P_s, k + 1, rm, i));
  float pz_k_c  = 0.5f * (pPad(p, P_t, P_s, k,     rc, i) + pPad(p, P_t, P_s, k + 1, rc, i));
  float pz_k1_m = 0.5f * (pPad(p, P_t, P_s, k + 1, rm, i) + pPad(p, P_t, P_s, k + 2, rm, i));
  float pz_k1_c = 0.5f * (pPad(p, P_t, P_s, k + 1, rc, i) + pPad(p, P_t, P_s, k + 2, rc, i));
  float dpzy = (0.5f * (pz_k1_m + pz_k1_c) - 0.5f * (pz_k_m + pz_k_c)) / DZf;

  float phz_m = 0.5f * (phiPad(Phi, Phi_t, Phi_s, k, rm, i) + phiPad(Phi, Phi_t, Phi_s, k + 1, rm, i));
  float phz_c = 0.5f * (phiPad(Phi, Phi_t, Phi_s, k, rc, i) + phiPad(Phi, Phi_t, Phi_s, k + 1, rc, i));
  float dphy = (phz_c - phz_m) / DYf;

  float R = term1 + term2 + term3 - muY * aY * dpy - dpzy * dphy;
  float coef = cfac * (float)dtp[0];
#if USE_ASYNC_LDS
  ASYNC_WAIT();
  float base = sBase[threadIdx.x];
#else
  float base = Vb[idx];
#endif
  Vout[idx] = base + coef * R;
}

// ---------------- kernel: R_W and W update ----------------
__global__ void k_rw(const float* __restrict__ Wb,
                     const float* __restrict__ U, const float* __restrict__ V,
                     const float* __restrict__ W, const float* __restrict__ Mu,
                     const float* __restrict__ alpha, const float* __restrict__ p,
                     const int* __restrict__ dtp, float cfac,
                     float* __restrict__ Wout) {
  int idx = blockIdx.x * blockDim.x + threadIdx.x;
  if (idx >= SZ_W) return;
#if USE_ASYNC_LDS
  __shared__ float sBase[BLK];
  ASYNC_STAGE(Wb + idx, &sBase[threadIdx.x]);
#endif
  int i = idx % GNX;
  int t = idx / GNX;
  int j = t % GNY;
  int k = t / GNY;   // k in [0,47)

  int xm = (i == 0) ? GNX - 1 : i - 1;
  int xp = (i == GNX - 1) ? 0 : i + 1;
  int jm = (j == 0) ? GNY - 1 : j - 1;
  int jp = (j == GNY - 1) ? 0 : j + 1;

  float w_c  = wVal(W, Mu, k, j, i);
  float w_xm = wVal(W, Mu, k, j, xm);
  float w_xp = wVal(W, Mu, k, j, xp);
  float w_ym = wVal(W, Mu, k, jm, i);
  float w_yp = wVal(W, Mu, k, jp, i);

  float Uz_i  = 0.5f * (U[idxU(k, j, i)]     + U[idxU(k + 1, j, i)]);
  float Uz_i1 = 0.5f * (U[idxU(k, j, i + 1)] + U[idxU(k + 1, j, i + 1)]);
  float term1 = -(Uz_i1 * (0.5f * (w_c + w_xp)) - Uz_i * (0.5f * (w_xm + w_c))) / DXf;

  float Vz_j  = 0.5f * (V[idxV(k, j, i)]     + V[idxV(k + 1, j, i)]);
  float Vz_j1 = 0.5f * (V[idxV(k, j + 1, i)] + V[idxV(k + 1, j + 1, i)]);
  float term2 = -(Vz_j1 * (0.5f * (w_c + w_yp)) - Vz_j * (0.5f * (w_ym + w_c))) / DYf;

  float Om_c  = OmegaVal(W, Mu, alpha, k, j, i);
  float Om_km = (k > 0) ? OmegaVal(W, Mu, alpha, k - 1, j, i) : 0.0f;
  float Om_kp = (k < GNZ - 2) ? OmegaVal(W, Mu, alpha, k + 1, j, i) : 0.0f;
  float w_km  = (k > 0) ? wVal(W, Mu, k - 1, j, i) : 0.0f;
  float w_kp  = (k < GNZ - 2) ? wVal(W, Mu, k + 1, j, i) : 0.0f;
  float term3 = -((0.5f * (Om_c + Om_kp)) * (0.5f * (w_c + w_kp)) -
                  (0.5f * (Om_km + Om_c)) * (0.5f * (w_km + w_c))) / DZf;

  float buoy = GGf * ((p[idxC(k + 1, j, i)] - p[idxC(k, j, i)]) / DZf -
                      0.5f * (Mu[idxC(k, j, i)] + Mu[idxC(k + 1, j, i)]));

  float coef = cfac * (float)dtp[0];
#if USE_ASYNC_LDS
  ASYNC_WAIT();
  float base = sBase[threadIdx.x];
#else
  float base = Wb[idx];
#endif
  Wout[idx] = base + coef * (term1 + term2 + term3 + buoy);
}

// ---------------- kernel: R_Theta + R_Mu, fused update ----------------
__global__ void k_rtm(const float* __restrict__ Tb, const float* __restrict__ Mb,
                      const float* __restrict__ U, const float* __restrict__ V,
                      const float* __restrict__ W, const float* __restrict__ Theta,
                      const float* __restrict__ Mu, const float* __restrict__ alpha,
                      const int* __restrict__ dtp, float cfac,
                      float* __restrict__ Tout, float* __restrict__ Mout) {
  int idx = blockIdx.x * blockDim.x + threadIdx.x;
  if (idx >= SZ_C) return;
#if USE_ASYNC_LDS
  __shared__ float sTb[BLK];
  __shared__ float sMb[BLK];
  ASYNC_STAGE(Tb + idx, &sTb[threadIdx.x]);
  ASYNC_STAGE(Mb + idx, &sMb[threadIdx.x]);
#endif
  int i = idx % GNX;
  int t = idx / GNX;
  int j = t % GNY;
  int k = t / GNY;

  int xm = (i == 0) ? GNX - 1 : i - 1;
  int xp = (i == GNX - 1) ? 0 : i + 1;
  int jm = (j == 0) ? GNY - 1 : j - 1;
  int jp = (j == GNY - 1) ? 0 : j + 1;

  float th_c  = Theta[idx] / Mu[idx];
  float th_xm = Theta[idxC(k, j, xm)] / Mu[idxC(k, j, xm)];
  float th_xp = Theta[idxC(k, j, xp)] / Mu[idxC(k, j, xp)];
  float th_ym = Theta[idxC(k, jm, i)] / Mu[idxC(k, jm, i)];
  float th_yp = Theta[idxC(k, jp, i)] / Mu[idxC(k, jp, i)];

  float Ui  = U[idxU(k, j, i)],  Ui1 = U[idxU(k, j, i + 1)];
  float Vj  = V[idxV(k, j, i)],  Vj1 = V[idxV(k, j + 1, i)];

  float t1 = -(Ui1 * (0.5f * (th_c + th_xp)) - Ui * (0.5f * (th_xm + th_c))) / DXf;
  float t2 = -(Vj1 * (0.5f * (th_c + th_yp)) - Vj * (0.5f * (th_ym + th_c))) / DYf;

  float OmP_k  = (k == 0) ? 0.0f : OmegaVal(W, Mu, alpha, k - 1, j, i);
  float OmP_k1 = (k == GNZ - 1) ? 0.0f : OmegaVal(W, Mu, alpha, k, j, i);
  float th_km = (k > 0) ? Theta[idxC(k - 1, j, i)] / Mu[idxC(k - 1, j, i)] : 0.0f;
  float th_kp = (k < GNZ - 1) ? Theta[idxC(k + 1, j, i)] / Mu[idxC(k + 1, j, i)] : 0.0f;
  float thz_k  = 0.5f * (th_km + th_c);
  float thz_k1 = 0.5f * (th_c + th_kp);
  float t3 = -(OmP_k1 * thz_k1 - OmP_k * thz_k) / DZf;

  float coef = cfac * (float)dtp[0];
  float rmu = -(Ui1 - Ui) / DXf - (Vj1 - Vj) / DYf - (OmP_k1 - OmP_k) / DZf;
#if USE_ASYNC_LDS
  ASYNC_WAIT();
  float baseT = sTb[threadIdx.x];
  float baseM = sMb[threadIdx.x];
#else
  float baseT = Tb[idx];
  float baseM = Mb[idx];
#endif
  Tout[idx] = baseT + coef * (t1 + t2 + t3);
  Mout[idx] = baseM + coef * rmu;
}

// ---------------- kernel: R_Phi and Phi update ----------------
__global__ void k_rphi(const float* __restrict__ Pb,
                       const float* __restrict__ U, const float* __restrict__ V,
                       const float* __restrict__ W, const float* __restrict__ Mu,
                       const float* __restrict__ Phi, const float* __restrict__ Phi_t,
                       const float* __restrict__ Phi_s,
                       const float* __restrict__ alpha,
                       const int* __restrict__ dtp, float cfac,
                       float* __restrict__ Pout) {
  int idx = blockIdx.x * blockDim.x + threadIdx.x;
  if (idx >= SZ_W) return;
#if USE_ASYNC_LDS
  __shared__ float sBase[BLK];
  ASYNC_STAGE(Pb + idx, &sBase[threadIdx.x]);
#endif
  int i = idx % GNX;
  int t = idx / GNX;
  int j = t % GNY;
  int k = t / GNY;   // k in [0,47)

  int xm = (i == 0) ? GNX - 1 : i - 1;
  int xp = (i == GNX - 1) ? 0 : i + 1;
  int jm = (j == 0) ? GNY - 1 : j - 1;
  int jp = (j == GNY - 1) ? 0 : j + 1;

  float uc_k  = 0.5f * (uFace(U, Mu, k,     j, i) + uFace(U, Mu, k,     j, i + 1));
  float uc_k1 = 0.5f * (uFace(U, Mu, k + 1, j, i) + uFace(U, Mu, k + 1, j, i + 1));
  float ucz = 0.5f * (uc_k + uc_k1);

  float vc_k  = 0.5f * (vFace(V, Mu, k,     j, i) + vFace(V, Mu, k,     j + 1, i));
  float vc_k1 = 0.5f * (vFace(V, Mu, k + 1, j, i) + vFace(V, Mu, k + 1, j + 1, i));
  float vcz = 0.5f * (vc_k + vc_k1);

  float dphix = (0.5f * (Phi[idxC(k, j, i)] + Phi[idxC(k, j, xp)]) -
                 0.5f * (Phi[idxC(k, j, xm)] + Phi[idxC(k, j, i)])) / DXf;
  float dphiy = (0.5f * (Phi[idxC(k, j, i)] + Phi[idxC(k, jp, i)]) -
                 0.5f * (Phi[idxC(k, jm, i)] + Phi[idxC(k, j, i)])) / DYf;

  float muz = 0.5f * (Mu[idxC(k, j, i)] + Mu[idxC(k + 1, j, i)]);
  float az  = 0.5f * (alpha[idxC(k, j, i)] + alpha[idxC(k + 1, j, i)]);
  float wv  = W[idxC(k, j, i)] / muz;
  float omega = -wv * GGf / az / muz;

  float phz_k  = 0.5f * (phiPad(Phi, Phi_t, Phi_s, k,     j, i) + phiPad(Phi, Phi_t, Phi_s, k + 1, j, i));
  float phz_k1 = 0.5f * (phiPad(Phi, Phi_t, Phi_s, k + 1, j, i) + phiPad(Phi, Phi_t, Phi_s, k + 2, j, i));
  float dphz = (phz_k1 - phz_k) / DZf;

  float R = -ucz * dphix - vcz * dphiy - omega * dphz + GGf * wv;
  float coef = cfac * (float)dtp[0];
#if USE_ASYNC_LDS
  ASYNC_WAIT();
  float base = sBase[threadIdx.x];
#else
  float base = Pb[idx];
#endif
  Pout[idx] = base + coef * R;
}

// ---------------- host launcher ----------------
extern "C" void kernel_launch(void* const* d_in, const int* in_sizes, int n_in,
                              void* d_out, int out_size, void* d_ws, size_t ws_size,
                              hipStream_t stream) {
  (void)in_sizes; (void)n_in; (void)out_size; (void)ws_size;

  const float* U0  = (const float*)d_in[0];
  const float* V0  = (const float*)d_in[1];
  const float* W0  = (const float*)d_in[2];
  const float* T0  = (const float*)d_in[3];
  const float* M0  = (const float*)d_in[4];
  const float* P0  = (const float*)d_in[5];
  const float* Pht = (const float*)d_in[6];
  const float* Phs = (const float*)d_in[7];
  const float* Ptp = (const float*)d_in[8];
  const float* Psp = (const float*)d_in[9];
  const int*   dtp = (const int*)d_in[10];

  float* out = (float*)d_out;
  float* oU = out;
  float* oV = oU + SZ_U;
  float* oW = oV + SZ_V;
  float* oT = oW + SZ_W;
  float* oM = oT + SZ_C;
  float* oP = oM + SZ_C;

  float* ws    = (float*)d_ws;
  float* alpha = ws;
  float* p     = alpha + SZ_C;
  float* bU    = p + SZ_C;
  float* bV    = bU + SZ_U;
  float* bW    = bV + SZ_V;
  float* bT    = bW + SZ_W;
  float* bM    = bT + SZ_C;
  float* bP    = bM + SZ_C;

  const int gC = (SZ_C + BLK - 1) / BLK;
  const int gU = (SZ_U + BLK - 1) / BLK;
  const int gV = (SZ_V + BLK - 1) / BLK;
  const int gW = (SZ_W + BLK - 1) / BLK;

  struct St  { const float *U, *V, *W, *T, *M, *P; };
  struct Ot  { float *U, *V, *W, *T, *M, *P; };
  St cur[3]  = { {U0, V0, W0, T0, M0, P0},
                 {oU, oV, oW, oT, oM, oP},
                 {bU, bV, bW, bT, bM, bP} };
  Ot outs[3] = { {oU, oV, oW, oT, oM, oP},
                 {bU, bV, bW, bT, bM, bP},
                 {oU, oV, oW, oT, oM, oP} };
  const float fac[3] = { 1.0f / 3.0f, 0.5f, 1.0f };

  for (int s = 0; s < 3; ++s) {
    const St& c = cur[s];
    const Ot& o = outs[s];
    k_derive<<<gC, BLK, 0, stream>>>(c.T, c.M, c.P, Pht, Phs, alpha, p);
    k_ru  <<<gU, BLK, 0, stream>>>(U0, c.U, c.V, c.W, c.M, c.P, Pht, Phs, Ptp, Psp,
                                   alpha, p, dtp, fac[s], o.U);
    k_rv  <<<gV, BLK, 0, stream>>>(V0, c.U, c.V, c.W, c.M, c.P, Pht, Phs, Ptp, Psp,
                                   alpha, p, dtp, fac[s], o.V);
    k_rw  <<<gW, BLK, 0, stream>>>(W0, c.U, c.V, c.W, c.M, alpha, p, dtp, fac[s], o.W);
    k_rtm <<<gC, BLK, 0, stream>>>(T0, M0, c.U, c.V, c.W, c.T, c.M, alpha, dtp, fac[s],
                                   o.T, o.M);
    k_rphi<<<gW, BLK, 0, stream>>>(P0, c.U, c.V, c.W, c.M, c.P, Pht, Phs, alpha, dtp,
                                   fac[s], o.P);
  }
}